// PointTransformerLayer_78357383348685
// MI455X (gfx1250) — compile-verified
//
#include <hip/hip_runtime.h>
#include <hip/hip_bf16.h>

// ---------------------------------------------------------------------------
// PointTransformerLayer — exact algebraic reduction.
//
// softmax over the neighbor axis j sums to 1, and v[b,i,c] is independent of
// j, so the whole pos-enc / energy MLP / softmax pipeline collapses exactly to
//        out = relu( x @ (wv @ wo) + bo ).
//
// Stage 1: Wt = (wv @ wo)^T      (64x64, stored transposed so stage-2 B
//                                 fragments are contiguous b64 loads)
// Stage 2: out = relu(X @ W + bo) as a WMMA fp32 GEMM:
//          M = B*N = 2048 rows, K = 64, N = 64
//          one wave32 per 16-row tile, 4x v8f accumulators (16x64 tile),
//          V_WMMA_F32_16X16X4_F32, 16 K-steps -> 64 wmma per wave.
// ---------------------------------------------------------------------------

typedef float v2f __attribute__((ext_vector_type(2)));
typedef float v8f __attribute__((ext_vector_type(8)));

#define PT_DIM 64

// ---- Stage 1: Wt[c][a] = sum_b wv[a][b] * wo[b][c]  (transposed store) -----
__global__ __launch_bounds__(256) void ptl_fuse_weights_t(
    const float* __restrict__ wv,   // [64,64] row-major
    const float* __restrict__ wo,   // [64,64] row-major
    float* __restrict__ Wt)         // [64,64] = (wv@wo)^T (workspace)
{
    const int idx = blockIdx.x * blockDim.x + threadIdx.x;   // 0..4095
    if (idx >= PT_DIM * PT_DIM) return;
    const int a = idx >> 6;          // K index (input dim of fused W)
    const int c = idx & 63;          // output column
    float s = 0.f;
#pragma unroll
    for (int b = 0; b < PT_DIM; ++b)
        s = fmaf(wv[a * PT_DIM + b], wo[b * PT_DIM + c], s);
    Wt[c * PT_DIM + a] = s;          // transposed: column-major fused W
}

// ---- Stage 2: out = relu(X @ W + bias), WMMA fp32 --------------------------
// Fragment layouts per CDNA5 ISA 7.12.2 (wave32):
//   A 16x4 f32 : lanes 0-15 -> M=lane, {v.x=K0, v.y=K1}
//                lanes 16-31 -> M=lane-16, {v.x=K2, v.y=K3}
//   B 4x16 f32 : mirrored K striping, N = lane&15 within each VGPR; with the
//                transposed Wt the two K elements per fragment are contiguous.
//   C/D 16x16  : VGPR r -> row (r + 8*(lane>=16)), col = lane&15
__global__ __launch_bounds__(256) void ptl_gemm_relu_wmma(
    const float* __restrict__ X,     // [M, 64]
    const float* __restrict__ Wt,    // [64, 64] = W^T (col-major W)
    const float* __restrict__ bias,  // [64]
    float* __restrict__ out,         // [M, 64]
    int M)
{
    const int lane  = threadIdx.x & 31;
    const int wave  = threadIdx.x >> 5;                  // 0..7
    const int tileM = (blockIdx.x * 8 + wave) * 16;      // 16-row output tile
    if (tileM >= M) return;                              // wave-uniform exit

    const int half = lane >> 4;       // 0: lanes 0-15, 1: lanes 16-31
    const int l15  = lane & 15;

    // Pull the 16 KB fused weight matrix toward the WGP early.
    __builtin_prefetch(Wt + lane * 32, 0, 3);

    v8f acc0 = {}, acc1 = {}, acc2 = {}, acc3 = {};

    const float* arow = X + (tileM + l15) * PT_DIM;
    // Column base pointers for this lane's four N-tiles (columns of W =
    // rows of Wt), each row of Wt is contiguous in K.
    const float* bc0 = Wt + (l15 +  0) * PT_DIM;
    const float* bc1 = Wt + (l15 + 16) * PT_DIM;
    const float* bc2 = Wt + (l15 + 32) * PT_DIM;
    const float* bc3 = Wt + (l15 + 48) * PT_DIM;

#pragma unroll
    for (int kk = 0; kk < PT_DIM; kk += 4) {
        const int k0 = kk + 2 * half;       // this lane's K pair
        // A fragment: contiguous pair -> one b64 load
        v2f a = *(const v2f*)(arow + k0);
        // B fragments: contiguous K pair in Wt -> one b64 load each
        v2f b0 = *(const v2f*)(bc0 + k0);
        v2f b1 = *(const v2f*)(bc1 + k0);
        v2f b2 = *(const v2f*)(bc2 + k0);
        v2f b3 = *(const v2f*)(bc3 + k0);

        acc0 = __builtin_amdgcn_wmma_f32_16x16x4_f32(false, a, false, b0,
                                                     (short)0, acc0, false, false);
        acc1 = __builtin_amdgcn_wmma_f32_16x16x4_f32(false, a, false, b1,
                                                     (short)0, acc1, false, false);
        acc2 = __builtin_amdgcn_wmma_f32_16x16x4_f32(false, a, false, b2,
                                                     (short)0, acc2, false, false);
        acc3 = __builtin_amdgcn_wmma_f32_16x16x4_f32(false, a, false, b3,
                                                     (short)0, acc3, false, false);
    }

    // Epilogue: bias + ReLU, C/D layout scatter.
    const int rowOff = tileM + 8 * half;
    const float bb0 = bias[l15 +  0];
    const float bb1 = bias[l15 + 16];
    const float bb2 = bias[l15 + 32];
    const float bb3 = bias[l15 + 48];
#pragma unroll
    for (int r = 0; r < 8; ++r) {
        float* orow = out + (rowOff + r) * PT_DIM + l15;
        float v0 = acc0[r] + bb0;
        float v1 = acc1[r] + bb1;
        float v2 = acc2[r] + bb2;
        float v3 = acc3[r] + bb3;
        orow[ 0] = v0 > 0.f ? v0 : 0.f;
        orow[16] = v1 > 0.f ? v1 : 0.f;
        orow[32] = v2 > 0.f ? v2 : 0.f;
        orow[48] = v3 > 0.f ? v3 : 0.f;
    }
}

extern "C" void kernel_launch(void* const* d_in, const int* in_sizes, int n_in,
                              void* d_out, int out_size, void* d_ws, size_t ws_size,
                              hipStream_t stream) {
    // setup_inputs order:
    //  0:x 1:pos 2:pe_w1 3:pe_b1 4:pe_w2 5:pe_b2 6:am_w1 7:am_b1 8:am_w2
    //  9:am_b2 10:wq 11:wk 12:wv 13:wo 14:bo
    const float* x  = (const float*)d_in[0];
    const float* wv = (const float*)d_in[12];
    const float* wo = (const float*)d_in[13];
    const float* bo = (const float*)d_in[14];
    float* out = (float*)d_out;

    const int M = in_sizes[0] / PT_DIM;     // B*N = 2048
    float* Wt = (float*)d_ws;               // 64*64 fp32 = 16 KB scratch

    // Stage 1: Wt = (wv @ wo)^T
    ptl_fuse_weights_t<<<(PT_DIM * PT_DIM + 255) / 256, 256, 0, stream>>>(wv, wo, Wt);

    // Stage 2: out = relu(x @ W + bo); 8 waves/block, 16-row tile per wave.
    const int tilesM = (M + 15) / 16;               // 128
    const int blocks = (tilesM + 7) / 8;            // 16
    ptl_gemm_relu_wmma<<<blocks, 256, 0, stream>>>(x, Wt, bo, out, M);
}